// DeformableConv2d_71141838291598
// MI455X (gfx1250) — compile-verified
//
#include <hip/hip_runtime.h>
#include <math.h>

// ---------------- problem constants ----------------
#define BATCH 4
#define CIN   64
#define HH    64
#define WW    256
#define COUT  128
#define KK    9            // 3x3 taps
#define CK    (CIN*KK)     // 576 reduction dim
#define HWSZ  (HH*WW)      // 16384 pixels per channel plane
#define OMCH  27           // offset/mask channels
#define TP    32           // pixel tile per workgroup (2 WMMA N-tiles)
#define NE    (TP*KK)      // 288 (tap,pixel) bilinear entries per block

typedef float v2f __attribute__((ext_vector_type(2)));
typedef float v8f __attribute__((ext_vector_type(8)));

// ============================================================
// Stage 1: offset/mask field  om[b][27][y][x] = conv3x3(x, ow) + ob
// (sigmoid on channels 18..26). Block = (b, oc, y), 256 thr = W row.
// ============================================================
__global__ __launch_bounds__(256) void offset_field_kernel(
    const float* __restrict__ x, const float* __restrict__ ow,
    const float* __restrict__ ob, float* __restrict__ om)
{
  const int xp = threadIdx.x;                 // 0..255
  const int y  = blockIdx.x & (HH - 1);
  const int oc = (blockIdx.x >> 6) % OMCH;
  const int b  = blockIdx.x / (HH * OMCH);

  const float* xb   = x  + (size_t)b * CIN * HWSZ;
  const float* wrow = ow + (size_t)oc * CIN * KK;

  float acc = ob[oc];
  for (int c = 0; c < CIN; ++c) {
    const float* xc = xb + c * HWSZ;
    #pragma unroll
    for (int t = 0; t < 9; ++t) {
      const int yy = y  - 1 + t / 3;
      const int xx = xp - 1 + t % 3;
      float v = 0.0f;
      if (yy >= 0 && yy < HH && xx >= 0 && xx < WW) v = xc[yy * WW + xx];
      acc = fmaf(v, wrow[c * KK + t], acc);
    }
  }
  if (oc >= 18) acc = 1.0f / (1.0f + __expf(-acc));   // mask channels
  om[((size_t)(b * OMCH + oc)) * HWSZ + y * WW + xp] = acc;
}

// ============================================================
// Stage 2: implicit GEMM with f32 WMMA 16x16x4.
//   D[128 oc][32 px] = W[128][576] * S[576][32]
//
// S layout in LDS (per k-quad of 128 words), chosen so that each
// WMMA B-fragment is ONE aligned ds_load_b64 and every read/write
// is bank-conflict-free:
//   pair = (ck>>1)&1
//   addr(ck,col) = (ck>>2)*128 + pair*64 + ((col ^ (pair<<4))<<1) + (ck&1)
// Reader (k-step kk, lane n, lane-half h):
//   tile0 frag @ word kk*128 + (h ? 96+2n : 2n)
//   tile1 frag @ word kk*128 + (h ? 64+2n : 32+2n)
// ============================================================
__device__ __forceinline__ int swz(int ck, int col) {
  const int pair = (ck >> 1) & 1;
  return ((ck >> 2) << 7) + (pair << 6) + ((col ^ (pair << 4)) << 1) + (ck & 1);
}

__global__ __launch_bounds__(256) void deform_wmma_kernel(
    const float* __restrict__ x, const float* __restrict__ om,
    const float* __restrict__ weight, float* __restrict__ out)
{
  __shared__ __align__(16) float sS[CK * TP];     // 73728 B, swizzled
  __shared__ int   sO00[NE], sO01[NE], sO10[NE], sO11[NE];  // clamped flat offs
  __shared__ float sM00[NE], sM01[NE], sM10[NE], sM11[NE];  // validity*mask*w

  const int t  = threadIdx.x;
  const int xt = blockIdx.x & 7;                  // 8 tiles of 32 px along W
  const int y  = (blockIdx.x >> 3) & (HH - 1);
  const int b  = blockIdx.x >> 9;
  const int x0 = xt * TP;

  // ---- Phase A.1: bilinear params, validity folded into weights ----
  for (int e = t; e < NE; e += 256) {
    const int p = e & (TP - 1), tap = e >> 5;     // tap 0..8
    const int xp = x0 + p;
    const float* omb = om + (size_t)b * OMCH * HWSZ + y * WW + xp;
    // interleaved: dy_k = om[2k], dx_k = om[2k+1], mask_k = om[18+k]
    const float dy = omb[(size_t)(2 * tap)     * HWSZ];
    const float dx = omb[(size_t)(2 * tap + 1) * HWSZ];
    const float m  = omb[(size_t)(18 + tap)    * HWSZ];
    const float py = (float)(y  - 1 + tap / 3) + dy;
    const float px = (float)(xp - 1 + tap % 3) + dx;
    const float y0f = floorf(py), x0f = floorf(px);
    const float ly = py - y0f, lx = px - x0f;
    const int y0 = (int)y0f, xi = (int)x0f;
    const int y0c = min(max(y0, 0), HH - 1),     y1c = min(max(y0 + 1, 0), HH - 1);
    const int x0c = min(max(xi, 0), WW - 1),     x1c = min(max(xi + 1, 0), WW - 1);
    const float vy0 = (y0 >= 0 && y0 < HH)         ? 1.0f : 0.0f;
    const float vy1 = (y0 + 1 >= 0 && y0 + 1 < HH) ? 1.0f : 0.0f;
    const float vx0 = (xi >= 0 && xi < WW)         ? 1.0f : 0.0f;
    const float vx1 = (xi + 1 >= 0 && xi + 1 < WW) ? 1.0f : 0.0f;
    sO00[e] = y0c * WW + x0c;  sM00[e] = (1.0f - ly) * (1.0f - lx) * m * vy0 * vx0;
    sO01[e] = y0c * WW + x1c;  sM01[e] = (1.0f - ly) * lx          * m * vy0 * vx1;
    sO10[e] = y1c * WW + x0c;  sM10[e] = ly * (1.0f - lx)          * m * vy1 * vx0;
    sO11[e] = y1c * WW + x1c;  sM11[e] = ly * lx                   * m * vy1 * vx1;
  }
  __syncthreads();

  // ---- Phase A.2: branchless gather-sample into swizzled S ----
  // tap outermost: read the 8 per-(tap,pixel) params ONCE into registers,
  // then the channel loop is pure global gathers + FMAs.
  {
    const int p = t & (TP - 1), g = t >> 5;       // g 0..7 -> 8 channels each
    const float* xb = x + (size_t)b * CIN * HWSZ + g * HWSZ;
    #pragma unroll
    for (int tap = 0; tap < 9; ++tap) {
      const int e = (tap << 5) | p;
      const int o00 = sO00[e], o01 = sO01[e], o10 = sO10[e], o11 = sO11[e];
      const float m00 = sM00[e], m01 = sM01[e], m10 = sM10[e], m11 = sM11[e];
      #pragma unroll
      for (int cc = 0; cc < 8; ++cc) {
        const float* xc = xb + (size_t)(cc << 3) * HWSZ;   // c = g + 8*cc
        float v;
        v = m00 * xc[o00];
        v = fmaf(m01, xc[o01], v);
        v = fmaf(m10, xc[o10], v);
        v = fmaf(m11, xc[o11], v);
        sS[swz((g + (cc << 3)) * KK + tap, p)] = v;
      }
    }
  }
  __syncthreads();

  // ---- Phase B: 8 waves; each = 16 oc x 32 px (two 16x16 D tiles) ----
  const int lane  = t & 31;
  const int wv    = t >> 5;
  const int oc0   = wv << 4;
  const int n     = lane & 15;
  const int khalf = (lane < 16) ? 0 : 2;          // A: lanes 16-31 hold K=2,3
  const int t0off = (lane < 16) ? (n << 1)        : 96 + (n << 1);  // tile0 frag
  const int t1off = (lane < 16) ? 32 + (n << 1)   : 64 + (n << 1);  // tile1 frag

  const float* arow = weight + (size_t)(oc0 + n) * CK + khalf;
  __builtin_prefetch(arow, 0, 1);                 // global_prefetch_b8

  v8f acc0 = {};                                  // cols x0 + 0..15
  v8f acc1 = {};                                  // cols x0 + 16..31
  #pragma unroll 4
  for (int kk = 0; kk < CK / 4; ++kk) {
    const v2f a  = *(const v2f*)(arow + kk * 4);           // global b64
    const float* sq = sS + (kk << 7);
    const v2f b0 = *(const v2f*)(sq + t0off);              // one ds_load_b64
    const v2f b1 = *(const v2f*)(sq + t1off);              // one ds_load_b64
    acc0 = __builtin_amdgcn_wmma_f32_16x16x4_f32(
        false, a, false, b0, (short)0, acc0, false, false);
    acc1 = __builtin_amdgcn_wmma_f32_16x16x4_f32(
        false, a, false, b1, (short)0, acc1, false, false);
  }

  // ---- store: C/D layout VGPR r -> M=r (lanes 0-15) / M=r+8 (lanes 16-31)
  const int mofs = (lane < 16) ? 0 : 8;
  float* ob = out + ((size_t)b * COUT + oc0) * HWSZ + y * WW + x0 + n;
  #pragma unroll
  for (int r = 0; r < 8; ++r) {
    ob[(size_t)(r + mofs) * HWSZ]      = acc0[r];
    ob[(size_t)(r + mofs) * HWSZ + 16] = acc1[r];
  }
}

extern "C" void kernel_launch(void* const* d_in, const int* in_sizes, int n_in,
                              void* d_out, int out_size, void* d_ws, size_t ws_size,
                              hipStream_t stream) {
  const float* x  = (const float*)d_in[0];   // [4,64,64,256]
  const float* ow = (const float*)d_in[1];   // [27,64,3,3]
  const float* ob = (const float*)d_in[2];   // [27]
  const float* w  = (const float*)d_in[3];   // [128,64,3,3]
  float* out = (float*)d_out;                // [4,128,64,256]
  float* om  = (float*)d_ws;                 // 4*27*16384*4 = 6.75 MB scratch

  offset_field_kernel<<<BATCH * OMCH * HH, 256, 0, stream>>>(x, ow, ob, om);
  deform_wmma_kernel<<<BATCH * HH * (WW / TP), 256, 0, stream>>>(x, om, w, out);
}